// BrainGNNSimplified_88785563943647
// MI455X (gfx1250) — compile-verified
//
#include <hip/hip_runtime.h>

#define NG      1024
#define NPG     116
#define NNODES  (NG*NPG)        // 118784
#define DEGK    32
#define EPG     (NPG*DEGK)      // 3712 edges per graph
#define NEDGES  (NNODES*DEGK)   // 3801088
#define INCH    116
#define HID     32
#define EPSV    1e-5f

typedef __attribute__((ext_vector_type(2))) float v2f;
typedef __attribute__((ext_vector_type(8))) float v8f;

// ---------------- utility ----------------
__global__ void k_fill(float* p, float v, int n) {
    int i = blockIdx.x * blockDim.x + threadIdx.x;
    if (i < n) p[i] = v;
}

// ---------------- fused degree -> dinv -> per-edge norm, one graph per block,
// all accumulation in LDS (ds_add_f32 atomics) ----------------
__global__ void k_degnorm(const int* __restrict__ row, const int* __restrict__ col,
                          const float* __restrict__ ew,
                          float* __restrict__ dinv, float* __restrict__ norm) {
    __shared__ float ldeg[NPG];
    __shared__ float ldinv[NPG];
    int g  = blockIdx.x;
    int n0 = g * NPG;
    int e0 = g * EPG;
    for (int i = threadIdx.x; i < NPG; i += blockDim.x) ldeg[i] = 1.0f;  // self loop
    __syncthreads();
    for (int t = threadIdx.x; t < EPG; t += blockDim.x)
        atomicAdd(&ldeg[col[e0 + t] - n0], ew[e0 + t]);
    __syncthreads();
    for (int i = threadIdx.x; i < NPG; i += blockDim.x) {
        float d = ldeg[i];
        float v = (d > 0.f) ? rsqrtf(d) : 0.f;
        ldinv[i] = v;
        dinv[n0 + i] = v;
    }
    __syncthreads();
    for (int t = threadIdx.x; t < EPG; t += blockDim.x) {
        int e = e0 + t;
        norm[e] = ldinv[row[e] - n0] * ew[e] * ldinv[col[e] - n0];
    }
}

// ---------------- WMMA GEMM: C[rows,NC] = A[rows,K] @ B[K,NC] ----------------
// rows % 16 == 0, K % 4 == 0, NC % 16 == 0. One wave per 16x16 tile, f32 WMMA.
__global__ void k_gemm_wmma(const float* __restrict__ A, const float* __restrict__ B,
                            float* __restrict__ C, int rows, int K, int NC) {
    int lane = threadIdx.x & 31;
    int wave = threadIdx.x >> 5;
    int colTiles = NC >> 4;
    int tile = blockIdx.x * (blockDim.x >> 5) + wave;
    int total = (rows >> 4) * colTiles;
    if (tile >= total) return;                 // wave-uniform: EXEC stays all-ones
    int r0 = (tile / colTiles) << 4;
    int c0 = (tile % colTiles) << 4;
    int half = lane >> 4;                      // 0: K pair {0,1}, 1: K pair {2,3}
    int l = lane & 15;                         // M (for A) / N (for B,C)
    const float* arow = A + (size_t)(r0 + l) * K;
    v8f acc = {0.f, 0.f, 0.f, 0.f, 0.f, 0.f, 0.f, 0.f};
    for (int k0 = 0; k0 < K; k0 += 4) {
        v2f a, b;
        a.x = arow[k0 + 2*half];
        a.y = arow[k0 + 2*half + 1];
        b.x = B[(size_t)(k0 + 2*half)     * NC + c0 + l];
        b.y = B[(size_t)(k0 + 2*half + 1) * NC + c0 + l];
        acc = __builtin_amdgcn_wmma_f32_16x16x4_f32(
            /*neg_a=*/false, a, /*neg_b=*/false, b,
            /*c_mod=*/(short)0, acc, /*reuse_a=*/false, /*reuse_b=*/false);
    }
#pragma unroll
    for (int r = 0; r < 8; ++r)
        C[(size_t)(r0 + r + 8*half) * NC + c0 + l] = acc[r];
}

// ---------------- graph-local scatter: one workgroup per graph ----------------
// LDS: hbuf (h tile), acc (aggregate, init = self-loop h*dinv^2), lst (BN stats).
// Edge aggregation uses LDS float atomics; BN col-sums fused into write-out.
__global__ void k_graph_scatter(const int* __restrict__ row, const int* __restrict__ col,
                                const float* __restrict__ norm,
                                const float* __restrict__ h, const float* __restrict__ dinv,
                                float* __restrict__ agg, float* __restrict__ stats,
                                int logC) {
    __shared__ float smem[NPG * 128 + 128];    // max C=64: 2 tiles + 2C stats
    int C = 1 << logC;
    int tile = NPG << logC;
    float* hbuf = smem;
    float* acc  = smem + tile;
    float* lst  = smem + 2 * tile;
    int g  = blockIdx.x;
    int n0 = g * NPG;
    int e0 = g * EPG;
    size_t base = (size_t)n0 << logC;

    // stage h tile + self-loop init
    for (int i = threadIdx.x; i < tile; i += blockDim.x) {
        float v = h[base + i];
        float d = dinv[n0 + (i >> logC)];
        hbuf[i] = v;
        acc[i]  = v * d * d;
    }
    if (threadIdx.x < 2 * C) lst[threadIdx.x] = 0.f;
    __syncthreads();

    // edge scatter in LDS: each thread handles one edge x 4 channels
    int itemsv = EPG << (logC - 2);
    int cmask = (C >> 2) - 1;
    for (int t = threadIdx.x; t < itemsv; t += blockDim.x) {
        int e  = e0 + (t >> (logC - 2));
        int c  = (t & cmask) << 2;
        int r  = (row[e] - n0) << logC;
        int cl = (col[e] - n0) << logC;
        float w = norm[e];
#pragma unroll
        for (int j = 0; j < 4; ++j)
            atomicAdd(&acc[cl + c + j], hbuf[r + c + j] * w);
    }
    __syncthreads();

    // write-out + fused BN column stats (partials in LDS, one global atomic/ch)
    for (int i = threadIdx.x; i < tile; i += blockDim.x) {
        float v = acc[i];
        agg[base + i] = v;
        int c = i & (C - 1);
        atomicAdd(&lst[c], v);
        atomicAdd(&lst[C + c], v * v);
    }
    __syncthreads();
    if (threadIdx.x < 2 * C) atomicAdd(&stats[threadIdx.x], lst[threadIdx.x]);
}

// stats[c] += sum, stats[C+c] += sumsq  (stats pre-zeroed) -- used for FC1 BN
__global__ void k_colstats(const float* __restrict__ h, float* __restrict__ stats,
                           int rows, int logC) {
    int t = blockIdx.x * blockDim.x + threadIdx.x;
    int C = 1 << logC;
    int c = t & (C - 1);
    int r = t >> logC;
    int rstride = (gridDim.x * blockDim.x) >> logC;
    float s = 0.f, q = 0.f;
    for (int i = r; i < rows; i += rstride) {
        float v = h[((size_t)i << logC) + c];
        s += v; q += v * v;
    }
    atomicAdd(&stats[c], s);
    atomicAdd(&stats[C + c], q);
}

// y = relu((x - mean) * rsqrt(var + eps) * g + be); biased stats from stats[]
__global__ void k_bnrelu(const float* __restrict__ x, const float* __restrict__ stats,
                         const float* __restrict__ g, const float* __restrict__ be,
                         float* __restrict__ y, int rows, int logC) {
    int i = blockIdx.x * blockDim.x + threadIdx.x;
    if (i < (rows << logC)) {
        int c = i & ((1 << logC) - 1);
        float inv_rows = 1.0f / (float)rows;
        float m = stats[c] * inv_rows;
        float var = stats[(1 << logC) + c] * inv_rows - m * m;
        float v = (x[i] - m) * rsqrtf(var + EPSV) * g[c] + be[c];
        y[i] = v > 0.f ? v : 0.f;
    }
}

// per-graph mean/max pool over 2*HID=64 channels -> emb[g, 128]
__global__ void k_pool(const float* __restrict__ h, float* __restrict__ emb) {
    int g = blockIdx.x;            // 1024 blocks
    int c = threadIdx.x;           // 64 threads
    const float* p = h + (size_t)g * NPG * 64 + c;
    float s = 0.f, mx = -INFINITY;
    for (int i = 0; i < NPG; ++i) {
        float v = p[(size_t)i * 64];
        s += v; mx = fmaxf(mx, v);
    }
    emb[g * 128 + c]      = s * (1.0f / (float)NPG);
    emb[g * 128 + 64 + c] = mx;
}

// out[g, j] = bf2[j] + sum_k z[g,k] * Wf2[k,j]   (j in {0,1})
__global__ void k_fc2(const float* __restrict__ z, const float* __restrict__ W,
                      const float* __restrict__ b, float* __restrict__ out) {
    int t = blockIdx.x * blockDim.x + threadIdx.x;
    if (t < NG * 2) {
        int g = t >> 1, j = t & 1;
        float s = b[j];
#pragma unroll
        for (int k = 0; k < HID; ++k) s += z[g * HID + k] * W[k * 2 + j];
        out[t] = s;
    }
}

static inline int cdiv(long long a, int b) { return (int)((a + b - 1) / b); }

extern "C" void kernel_launch(void* const* d_in, const int* in_sizes, int n_in,
                              void* d_out, int out_size, void* d_ws, size_t ws_size,
                              hipStream_t stream) {
    const float* x   = (const float*)d_in[0];
    const int*   ei  = (const int*)d_in[1];   // [2, E] int32 (JAX x64 disabled)
    const float* ew  = (const float*)d_in[2];
    // d_in[3] batch: implicit (nodes contiguous per graph)
    const float* W1  = (const float*)d_in[4];
    // b1 (d_in[5]) cancels in BN
    const float* g1  = (const float*)d_in[6];
    const float* be1 = (const float*)d_in[7];
    const float* W2  = (const float*)d_in[8];
    // b2 cancels
    const float* g2  = (const float*)d_in[10];
    const float* be2 = (const float*)d_in[11];
    const float* Wf1 = (const float*)d_in[12];
    // bf1 cancels
    const float* gf  = (const float*)d_in[14];
    const float* bef = (const float*)d_in[15];
    const float* Wf2 = (const float*)d_in[16];
    const float* bf2 = (const float*)d_in[17];
    const int* row = ei;
    const int* col = ei + NEDGES;

    float* ws    = (float*)d_ws;
    float* dinv  = ws;                                  // N
    float* norm  = dinv + NNODES;                       // E
    float* h1    = norm + NEDGES;                       // N*32
    float* agg1  = h1   + (size_t)NNODES * 32;          // N*32
    float* h2    = agg1 + (size_t)NNODES * 32;          // N*64
    float* agg2  = h2   + (size_t)NNODES * 64;          // N*64
    float* emb   = agg2 + (size_t)NNODES * 64;          // 1024*128
    float* z1    = emb  + NG * 128;                     // 1024*32
    float* stats = z1   + NG * HID;                     // 128

    const int B = 256;

    // fused degree (self-loop 1.0) -> rsqrt -> per-edge sym-norm, LDS atomics
    k_degnorm<<<NG, B, 0, stream>>>(row, col, ew, dinv, norm);

    // ---- conv1: h1 = x @ W1 ; LDS graph scatter (+self-loop,+BN stats) ; BN+ReLU -> h1 ----
    {
        int tiles = (NNODES / 16) * (HID / 16);
        k_gemm_wmma<<<cdiv(tiles, B / 32), B, 0, stream>>>(x, W1, h1, NNODES, INCH, HID);
    }
    k_fill<<<1, 64, 0, stream>>>(stats, 0.f, 64);
    k_graph_scatter<<<NG, B, 0, stream>>>(row, col, norm, h1, dinv, agg1, stats, 5);
    k_bnrelu<<<cdiv((long long)NNODES * 32, B), B, 0, stream>>>(agg1, stats, g1, be1, h1, NNODES, 5);

    // ---- conv2: h2 = h1 @ W2 ; LDS graph scatter ; BN+ReLU -> h2 ----
    {
        int tiles = (NNODES / 16) * (2 * HID / 16);
        k_gemm_wmma<<<cdiv(tiles, B / 32), B, 0, stream>>>(h1, W2, h2, NNODES, HID, 2 * HID);
    }
    k_fill<<<1, 128, 0, stream>>>(stats, 0.f, 128);
    k_graph_scatter<<<NG, B, 0, stream>>>(row, col, norm, h2, dinv, agg2, stats, 6);
    k_bnrelu<<<cdiv((long long)NNODES * 64, B), B, 0, stream>>>(agg2, stats, g2, be2, h2, NNODES, 6);

    // ---- pool -> emb[1024,128] ----
    k_pool<<<NG, 64, 0, stream>>>(h2, emb);

    // ---- fc1: z1 = emb @ Wf1 ; BN+ReLU (in place) ----
    {
        int tiles = (NG / 16) * (HID / 16);
        k_gemm_wmma<<<cdiv(tiles, B / 32), B, 0, stream>>>(emb, Wf1, z1, NG, 4 * HID, HID);
    }
    k_fill<<<1, 64, 0, stream>>>(stats, 0.f, 64);
    k_colstats<<<256, B, 0, stream>>>(z1, stats, NG, 5);
    k_bnrelu<<<cdiv((long long)NG * HID, B), B, 0, stream>>>(z1, stats, gf, bef, z1, NG, 5);

    // ---- fc2 -> out[1024,2] ----
    k_fc2<<<cdiv(NG * 2, B), B, 0, stream>>>(z1, Wf2, bf2, (float*)d_out);
}